// SemanticModule_81166291960305
// MI455X (gfx1250) — compile-verified
//
#include <hip/hip_runtime.h>

// ---------------------------------------------------------------------------
// Hetero GraphConv (7 layers, 5 relations) for MI455X / gfx1250.
// Phase 1 (dominant): edge scatter, bandwidth/atomic bound; agg buffers are
//   51-102MB so fp32 atomics resolve in the 192MB L2.
// Phase 2: per-destination fully-fused layer kernel -- all 5 relation GEMMs
//   (exact fp32 WMMA, V_WMMA_F32_16X16X4_F32) accumulate one C tile in
//   registers, epilogue applies bias + residual + relu. One read/write pass
//   over the output instead of seven.
// ---------------------------------------------------------------------------

#define NS_ 400000   // stroke nodes (multiple of 16)
#define NL_ 200000   // loop nodes   (multiple of 16)
#define NE_ 1600000  // edges per relation
#define LD  64       // padded feature stride

typedef float v2f __attribute__((ext_vector_type(2)));
typedef float v8f __attribute__((ext_vector_type(8)));

// ---- elementwise helpers ---------------------------------------------------

__global__ void k_fill04(float4* __restrict__ p, int n4) {
  int t = blockIdx.x * blockDim.x + threadIdx.x;
  if (t < n4) p[t] = make_float4(0.f, 0.f, 0.f, 0.f);
}

// src: n x cin dense -> dst: n x LD zero-padded
__global__ void k_pad(const float* __restrict__ src, int cin,
                      float* __restrict__ dst, int n) {
  int t = blockIdx.x * blockDim.x + threadIdx.x;
  if (t >= n * LD) return;
  int col = t & (LD - 1);
  int row = t >> 6;
  dst[t] = (col < cin) ? src[row * cin + col] : 0.0f;
}

// o = a + b + c + d (combine root weights / biases of the 4 loop relations)
__global__ void k_sum4(const float* __restrict__ a, const float* __restrict__ b,
                       const float* __restrict__ c, const float* __restrict__ d,
                       float* __restrict__ o, int n) {
  int t = blockIdx.x * blockDim.x + threadIdx.x;
  if (t < n) o[t] = a[t] + b[t] + c[t] + d[t];
}

// in-degree count per destination node
__global__ void k_count(const int* __restrict__ ei, float* __restrict__ cnt) {
  int e = blockIdx.x * blockDim.x + threadIdx.x;
  if (e >= NE_) return;
  unsafeAtomicAdd(&cnt[ei[NE_ + e]], 1.0f);
}

__global__ void k_recip(float* __restrict__ r, int n) {
  int t = blockIdx.x * blockDim.x + threadIdx.x;
  if (t < n) r[t] = 1.0f / fmaxf(r[t], 1.0f);
}

// AGG[dst] += X[src]; one thread per (edge, 4-feature chunk):
// one b128 load + 4 fp32 atomics that resolve in L2.
__global__ void k_scatter4(const int* __restrict__ ei, const float* __restrict__ X,
                           float* __restrict__ AGG, int chunks) {
  int t = blockIdx.x * blockDim.x + threadIdx.x;
  if (t >= NE_ * chunks) return;
  int e = t / chunks;
  int f0 = (t - e * chunks) << 2;
  int s = ei[e];
  int d = ei[NE_ + e];
  const float4 v = *(const float4*)(X + s * LD + f0);
  float* p = AGG + d * LD + f0;
  unsafeAtomicAdd(p + 0, v.x);
  unsafeAtomicAdd(p + 1, v.y);
  unsafeAtomicAdd(p + 2, v.z);
  unsafeAtomicAdd(p + 3, v.w);
}

// ---- WMMA building block ---------------------------------------------------
// Accumulate C += scale(A_row_frag) @ W over K (A frag 16x4: lanes 0-15 hold
// K=k0,k0+1; lanes 16-31 K=k0+2,k0+3 -> aligned b64 load. B frag 4x16
// mirrored by column). Ar is pre-offset by row*LD + hi*2; Wc by col + hi*2*COUT.
template <int K, int KCAP, int COUTT, bool SCALED>
__device__ inline v8f mmK(const float* __restrict__ Ar, const float* __restrict__ Wc,
                          float sc, int hi, v8f c) {
#pragma unroll
  for (int k0 = 0; k0 < K; k0 += 4) {
    const int ka = k0 + 2 * hi;  // lane-half K base (KCAP guard, layer 0 only)
    v2f a = *(const v2f*)(Ar + k0);
    if (SCALED) { a.x *= sc; a.y *= sc; }
    v2f b;
    b.x = (KCAP == K || ka < KCAP)     ? Wc[(size_t)k0 * COUTT]       : 0.0f;
    b.y = (KCAP == K || ka + 1 < KCAP) ? Wc[(size_t)(k0 + 1) * COUTT] : 0.0f;
    c = __builtin_amdgcn_wmma_f32_16x16x4_f32(false, a, false, b,
                                              (short)0, c, false, false);
  }
  return c;
}

// ---- fused per-layer kernels -----------------------------------------------
// One wave per 16x16 output tile; blockDim = 32*(COUT/16) -> EXEC all-ones at
// every WMMA. C stays in registers across all relation GEMMs; epilogue does
// bias + (residual) + (relu) in a single output pass. nrows % 16 == 0.

struct LoopArgs {   // destination: loop  (rb sum, nv/nh/ct mean, + 4 roots)
  const float *X, *Arb, *Anv, *Anh, *Act;
  const float *rnv, *rnh, *rct;
  const float *Wrt, *brt, *Wrb, *Wnv, *Wnh, *Wct;
  float* OUT;
};
struct StrokeArgs { // destination: stroke (od sum + root)
  const float *X, *Aod;
  const float *Wrt, *b, *Wod;
  float* OUT;
};

template <int K, int KCAP, int COUTT, bool RELU, bool RES>
__global__ void __launch_bounds__(32 * (COUTT / 16))
k_loop_layer(LoopArgs p) {
  const int lane = threadIdx.x & 31;
  const int wave = threadIdx.x >> 5;
  const int row0 = blockIdx.x << 4;
  const int l15 = lane & 15;
  const int hi  = lane >> 4;
  const size_t aro = (size_t)(row0 + l15) * LD + hi * 2;
  const int col = (wave << 4) + l15;
  const size_t wco = (size_t)col + (size_t)hi * 2 * COUTT;

  const float snv = p.rnv[row0 + l15];
  const float snh = p.rnh[row0 + l15];
  const float sct = p.rct[row0 + l15];

  v8f c = {};
  c = mmK<K, KCAP, COUTT, false>(p.X   + aro, p.Wrt + wco, 0.f, hi, c);
  c = mmK<K, KCAP, COUTT, false>(p.Arb + aro, p.Wrb + wco, 0.f, hi, c);
  c = mmK<K, KCAP, COUTT, true >(p.Anv + aro, p.Wnv + wco, snv, hi, c);
  c = mmK<K, KCAP, COUTT, true >(p.Anh + aro, p.Wnh + wco, snh, hi, c);
  c = mmK<K, KCAP, COUTT, true >(p.Act + aro, p.Wct + wco, sct, hi, c);

  const float bias = p.brt[col];
#pragma unroll
  for (int g = 0; g < 8; ++g) {
    size_t i = (size_t)(row0 + g + hi * 8) * LD + col;
    float v = c[g] + bias;
    if (RES) v += p.X[i];
    if (RELU) v = fmaxf(v, 0.f);
    p.OUT[i] = v;
  }
}

template <int K, int KCAP, int COUTT, bool RELU, bool RES>
__global__ void __launch_bounds__(32 * (COUTT / 16))
k_stroke_layer(StrokeArgs p) {
  const int lane = threadIdx.x & 31;
  const int wave = threadIdx.x >> 5;
  const int row0 = blockIdx.x << 4;
  const int l15 = lane & 15;
  const int hi  = lane >> 4;
  const size_t aro = (size_t)(row0 + l15) * LD + hi * 2;
  const int col = (wave << 4) + l15;
  const size_t wco = (size_t)col + (size_t)hi * 2 * COUTT;

  v8f c = {};
  c = mmK<K, KCAP, COUTT, false>(p.X   + aro, p.Wrt + wco, 0.f, hi, c);
  c = mmK<K, KCAP, COUTT, false>(p.Aod + aro, p.Wod + wco, 0.f, hi, c);

  const float bias = p.b[col];
#pragma unroll
  for (int g = 0; g < 8; ++g) {
    size_t i = (size_t)(row0 + g + hi * 8) * LD + col;
    float v = c[g] + bias;
    if (RES) v += p.X[i];
    if (RELU) v = fmaxf(v, 0.f);
    p.OUT[i] = v;
  }
}

// ---- host orchestration ----------------------------------------------------

static inline unsigned cdiv(long long a, long long b) { return (unsigned)((a + b - 1) / b); }

extern "C" void kernel_launch(void* const* d_in, const int* in_sizes, int n_in,
                              void* d_out, int out_size, void* d_ws, size_t ws_size,
                              hipStream_t stream) {
  const float* x_stroke = (const float*)d_in[0];
  const float* x_loop   = (const float*)d_in[1];
  const int*   e_rb     = (const int*)d_in[2];
  const int*   e_nv     = (const int*)d_in[3];
  const int*   e_nh     = (const int*)d_in[4];
  const int*   e_ct     = (const int*)d_in[5];
  const int*   e_od     = (const int*)d_in[6];
  const float* params   = (const float*)d_in[7];
  float* out = (float*)d_out;

  static const int CIN_[7]  = {6, 16, 32, 64, 64, 64, 64};
  static const int COUT_[7] = {16, 32, 64, 64, 64, 64, 64};
  // params flatten order within a layer (sorted dict keys): ct, nh, nv, od, rb
  const int CT = 0, NH = 1, NV = 2, OD = 3, RB = 4;

  // workspace carve-out
  char* ws = (char*)d_ws;
  size_t off = 0;
  auto carve = [&](size_t nfloats) {
    float* p = (float*)(ws + off);
    off = (off + nfloats * sizeof(float) + 255) & ~(size_t)255;
    return p;
  };
  float* XS[2]  = {carve((size_t)NS_ * LD), carve((size_t)NS_ * LD)};
  float* XL[2]  = {carve((size_t)NL_ * LD), carve((size_t)NL_ * LD)};
  float* A_OD   = carve((size_t)NS_ * LD);
  float* A_RB   = carve((size_t)NL_ * LD);
  float* A_NV   = carve((size_t)NL_ * LD);
  float* A_NH   = carve((size_t)NL_ * LD);
  float* A_CT   = carve((size_t)NL_ * LD);
  float* RC[3]  = {carve(NL_), carve(NL_), carve(NL_)};  // recips nv, nh, ct
  float* WCMB   = carve(64 * 64);                        // combined loop root W
  float* BCMB   = carve(64);                             // combined loop root b

  const int TB = 256;

  // pad inputs into stride-64 buffers (zero pad -> K rounded to mult of 4 safe)
  k_pad<<<cdiv((long long)NS_ * LD, TB), TB, 0, stream>>>(x_stroke, 6, XS[0], NS_);
  k_pad<<<cdiv((long long)NL_ * LD, TB), TB, 0, stream>>>(x_loop,   6, XL[0], NL_);

  // per-relation mean reciprocals (edges constant across layers)
  const int* mean_e[3] = {e_nv, e_nh, e_ct};
  for (int i = 0; i < 3; ++i) {
    k_fill04<<<cdiv(NL_ / 4, TB), TB, 0, stream>>>((float4*)RC[i], NL_ / 4);
    k_count<<<cdiv(NE_, TB), TB, 0, stream>>>(mean_e[i], RC[i]);
    k_recip<<<cdiv(NL_, TB), TB, 0, stream>>>(RC[i], NL_);
  }

  size_t poff = 0;
  int cur = 0;
  for (int l = 0; l < 7; ++l) {
    const int cin = CIN_[l], cout = COUT_[l];
    const int chunks = ((cin + 3) & ~3) / 4;
    const int nxt = 1 - cur;

    const size_t relsz = (size_t)2 * cin * cout + cout;
    const float* P = params + poff;
    poff += 5 * relsz;
    auto Wrel  = [&](int r) { return P + r * relsz; };
    auto Brel  = [&](int r) { return P + r * relsz + (size_t)cin * cout; };
    auto Wroot = [&](int r) { return P + r * relsz + (size_t)cin * cout + cout; };

    // ---- scatter phase: all 5 relation aggregations ----
    struct { const int* e; const float* x; float* agg; int n; } sc[5] = {
      {e_od, XS[cur], A_OD, NS_}, {e_rb, XS[cur], A_RB, NL_},
      {e_nv, XL[cur], A_NV, NL_}, {e_nh, XL[cur], A_NH, NL_},
      {e_ct, XL[cur], A_CT, NL_}};
    for (int i = 0; i < 5; ++i) {
      k_fill04<<<cdiv((long long)sc[i].n * LD / 4, TB), TB, 0, stream>>>(
          (float4*)sc[i].agg, sc[i].n * LD / 4);
      k_scatter4<<<cdiv((long long)NE_ * chunks, TB), TB, 0, stream>>>(
          sc[i].e, sc[i].x, sc[i].agg, chunks);
    }

    // combined loop root weight/bias (HeteroConv sum over 4 relations)
    k_sum4<<<cdiv(cin * cout, TB), TB, 0, stream>>>(Wroot(CT), Wroot(NH), Wroot(NV),
                                                    Wroot(RB), WCMB, cin * cout);
    k_sum4<<<1, 64, 0, stream>>>(Brel(CT), Brel(NH), Brel(NV), Brel(RB), BCMB, cout);

    // ---- fused GEMM + bias + residual + relu ----
    LoopArgs la = {XL[cur], A_RB, A_NV, A_NH, A_CT, RC[0], RC[1], RC[2],
                   WCMB, BCMB, Wrel(RB), Wrel(NV), Wrel(NH), Wrel(CT), XL[nxt]};
    StrokeArgs sa = {XS[cur], A_OD, Wroot(OD), Brel(OD), Wrel(OD), XS[nxt]};
    dim3 gL(NL_ / 16), gS(NS_ / 16);
    const int bt = 32 * (cout / 16);
    switch (l) {
      case 0:  // head: no relu, no residual
        k_loop_layer  <8, 6, 16, false, false><<<gL, bt, 0, stream>>>(la);
        k_stroke_layer<8, 6, 16, false, false><<<gS, bt, 0, stream>>>(sa);
        break;
      case 1:  // 16->32: relu, no residual
        k_loop_layer  <16, 16, 32, true, false><<<gL, bt, 0, stream>>>(la);
        k_stroke_layer<16, 16, 32, true, false><<<gS, bt, 0, stream>>>(sa);
        break;
      case 2:  // 32->64: relu, no residual
        k_loop_layer  <32, 32, 64, true, false><<<gL, bt, 0, stream>>>(la);
        k_stroke_layer<32, 32, 64, true, false><<<gS, bt, 0, stream>>>(sa);
        break;
      default: // 64->64: relu + residual
        k_loop_layer  <64, 64, 64, true, true><<<gL, bt, 0, stream>>>(la);
        k_stroke_layer<64, 64, 64, true, true><<<gS, bt, 0, stream>>>(sa);
        break;
    }
    cur = nxt;
  }

  // final relu is idempotent (layer-6 output already relu'd); stride==cout==64
  hipMemcpyAsync(out, XS[cur], (size_t)NS_ * LD * sizeof(float),
                 hipMemcpyDeviceToDevice, stream);
  hipMemcpyAsync(out + (size_t)NS_ * LD, XL[cur], (size_t)NL_ * LD * sizeof(float),
                 hipMemcpyDeviceToDevice, stream);
}